// FeedForwardQuantum_65481071398177
// MI455X (gfx1250) — compile-verified
//
#include <hip/hip_runtime.h>

typedef float v2f __attribute__((ext_vector_type(2)));
typedef float v4f __attribute__((ext_vector_type(4)));
typedef float v8f __attribute__((ext_vector_type(8)));

#define E 8
#define F 32
#define WAVES_PER_BLOCK 8
#define HPITCH 34  // padded DWORD pitch for the 16x32 h tile (bank-conflict free, keeps float2 8B-aligned)

__launch_bounds__(256, 1)
__global__ void ffq_kernel(const float* __restrict__ x,
                           const float* __restrict__ theta,
                           const float* __restrict__ w1,
                           const float* __restrict__ b1,
                           const float* __restrict__ w2,
                           const float* __restrict__ b2,
                           float* __restrict__ out,
                           int n_tiles) {
  __shared__ float lds_h[WAVES_PER_BLOCK][16 * HPITCH];
  __shared__ float lds_o[WAVES_PER_BLOCK][128];

  const int tid  = threadIdx.x;
  const int wave = tid >> 5;
  const int lane = tid & 31;
  const int lo   = lane & 15;   // M (A/C col-lane) index
  const int hi   = lane >> 4;   // half-wave select
  const int c    = hi * 2;      // K offset inside a 4-wide K chunk

  // ---------- tile-invariant operand preload ----------
  const float t0 = theta[c], t1 = theta[c + 1], t2 = theta[c + 4], t3 = theta[c + 5];

  // GEMM1 B operands: w1 is [E=8][F=32] row-major; chunk k covers K rows 4k..4k+3.
  // B 4x16 layout: lane = N; lanes 0-15 hold K=4k,4k+1; lanes 16-31 hold K=4k+2,4k+3.
  v2f bw1[2][2];
  for (int nt = 0; nt < 2; ++nt) {
    const int n = nt * 16 + lo;
    for (int k = 0; k < 2; ++k) {
      bw1[nt][k].x = w1[(4 * k + c    ) * F + n];
      bw1[nt][k].y = w1[(4 * k + c + 1) * F + n];
    }
  }
  const float bias1_0 = b1[lo];
  const float bias1_1 = b1[16 + lo];

  // GEMM2 B operands: w2 is [F=32][E=8]; pad output cols 8..15 with zeros.
  v2f bw2[8];
  for (int k = 0; k < 8; ++k) {
    if (lo < E) {
      bw2[k].x = w2[(4 * k + c    ) * E + lo];
      bw2[k].y = w2[(4 * k + c + 1) * E + lo];
    } else {
      bw2[k].x = 0.0f;
      bw2[k].y = 0.0f;
    }
  }
  const float bias2 = (lo < E) ? b2[lo] : 0.0f;

  float* hbuf = lds_h[wave];
  float* obuf = lds_o[wave];

  const int wave_gid = blockIdx.x * WAVES_PER_BLOCK + wave;
  const int n_waves  = gridDim.x * WAVES_PER_BLOCK;

  for (int tile = wave_gid; tile < n_tiles; tile += n_waves) {
    // ---- stream 16 rows of x; both half-waves together read each row exactly once ----
    const int row = tile * 16 + lo;
    const float* px = x + row * E + c;
    const v2f xa = *(const v2f*)(px);       // elements c,   c+1
    const v2f xb = *(const v2f*)(px + 4);   // elements c+4, c+5

    v2f A0, A1;
    A0.x = __builtin_cosf(xa.x + t0);
    A0.y = __builtin_cosf(xa.y + t1);
    A1.x = __builtin_cosf(xb.x + t2);
    A1.y = __builtin_cosf(xb.y + t3);

    // ---- GEMM1: h(16x32) = A(16x8) @ w1(8x32), two N-tiles, K in two chunks ----
    v8f acc0 = {};
    v8f acc1 = {};
    acc0 = __builtin_amdgcn_wmma_f32_16x16x4_f32(false, A0, false, bw1[0][0], (short)0, acc0, false, false);
    acc0 = __builtin_amdgcn_wmma_f32_16x16x4_f32(false, A1, false, bw1[0][1], (short)0, acc0, false, false);
    acc1 = __builtin_amdgcn_wmma_f32_16x16x4_f32(false, A0, false, bw1[1][0], (short)0, acc1, false, false);
    acc1 = __builtin_amdgcn_wmma_f32_16x16x4_f32(false, A1, false, bw1[1][1], (short)0, acc1, false, false);

    // ---- bias + ReLU; stash h to LDS (C layout: VGPR r = row 8*hi+r, lane = col) ----
#pragma unroll
    for (int r = 0; r < 8; ++r) {
      float h0 = acc0[r] + bias1_0; h0 = h0 > 0.0f ? h0 : 0.0f;
      float h1 = acc1[r] + bias1_1; h1 = h1 > 0.0f ? h1 : 0.0f;
      hbuf[(8 * hi + r) * HPITCH + lo     ] = h0;
      hbuf[(8 * hi + r) * HPITCH + 16 + lo] = h1;
    }

    // ---- GEMM2: out(16x8) = h(16x32) @ w2(32x8); K=32 -> 8 chained f32 WMMAs ----
    v8f acc2 = {};
#pragma unroll
    for (int k = 0; k < 8; ++k) {
      // A-layout reload: row = lo, K = 4k+c, 4k+c+1 (8B-aligned thanks to HPITCH=34)
      const v2f a = *(const v2f*)&hbuf[lo * HPITCH + 4 * k + c];
      acc2 = __builtin_amdgcn_wmma_f32_16x16x4_f32(false, a, false, bw2[k], (short)0, acc2, false, false);
    }

    // ---- bias2, stage 16x8 tile (128 contiguous floats) through LDS ----
    if (lo < E) {
#pragma unroll
      for (int r = 0; r < 8; ++r)
        obuf[(8 * hi + r) * E + lo] = acc2[r] + bias2;
    }

    // fully coalesced 512B wave store
    const v4f o = *(const v4f*)&obuf[lane * 4];
    *(v4f*)(out + tile * 128 + lane * 4) = o;
  }
}

extern "C" void kernel_launch(void* const* d_in, const int* in_sizes, int n_in,
                              void* d_out, int out_size, void* d_ws, size_t ws_size,
                              hipStream_t stream) {
  const float* x     = (const float*)d_in[0];
  const float* theta = (const float*)d_in[1];
  const float* w1    = (const float*)d_in[2];
  const float* b1    = (const float*)d_in[3];
  const float* w2    = (const float*)d_in[4];
  const float* b2    = (const float*)d_in[5];
  float* out = (float*)d_out;

  const int n_rows  = in_sizes[0] / E;   // B*S = 524288
  const int n_tiles = n_rows / 16;       // 32768 (exact)

  ffq_kernel<<<1024, 256, 0, stream>>>(x, theta, w1, b1, w2, b2, out, n_tiles);
}